// IndexWindowMemoryEntityModel_56272661512411
// MI455X (gfx1250) — compile-verified
//
#include <hip/hip_runtime.h>
#include <hip/hip_bf16.h>
#include <math.h>

typedef __attribute__((ext_vector_type(2))) float v2f;
typedef __attribute__((ext_vector_type(8))) float v8f;

#define M_TILE 64
#define N_TILE 128
#define K_CHUNK 16
#define H_DIM 2048
#define T_LEN 512
#define E_DIM 300
#define K_WIN 1500   // 5 * 300
#define NUM_KC 94    // ceil(1500 / 16)

// ---------------------------------------------------------------------------
// Kernel 1: Hpre[(b,t), h] = sum_{k<1500} A[(b,t),k] * W1[k,h]
// A[(b,t), kw*300+e] = emb_table[ idxs_pad[b, t+kw] , e ]   (pad token = 0)
// fp32 WMMA 16x16x4. Block: 256 thr (8 waves), 64x128 output tile.
// Staging is software-pipelined through registers: global loads for chunk
// kc+1 are issued before the WMMA compute of chunk kc, so load latency
// overlaps matrix math instead of serializing into load->wait->ds_store.
// ---------------------------------------------------------------------------
__global__ __launch_bounds__(256)
void winproj_gemm_f32(const int* __restrict__ idxs,
                      const float* __restrict__ emb,
                      const float* __restrict__ W1,
                      float* __restrict__ Hpre) {
    __shared__ float Alds[M_TILE * 17];          // padded stride 17
    __shared__ float Blds[K_CHUNK * N_TILE];
    __shared__ int   tokLds[M_TILE * 5];

    const int tid   = threadIdx.x;
    const int lane  = tid & 31;
    const int wave  = tid >> 5;
    const int half  = lane >> 4;                 // 0 -> K{0,1}/M{0..7}, 1 -> K{2,3}/M{8..15}
    const int l16   = lane & 15;
    const int waveM = wave & 1;                  // 2 M-subtiles of 32
    const int waveN = wave >> 1;                 // 4 N-subtiles of 32
    const int mBlk  = blockIdx.x * M_TILE;
    const int nBlk  = blockIdx.y * N_TILE;

    // Stage the 5 window tokens for each of the 64 rows of this block.
    if (tid < M_TILE * 5) {
        int r  = tid / 5;
        int kw = tid - r * 5;
        int m  = mBlk + r;
        int b  = m >> 9;               // T = 512
        int t  = m & 511;
        int oi = t + kw - 2;           // WINDOW_PAD = 2; pad index -> token 0
        tokLds[tid] = (oi >= 0 && oi < T_LEN) ? idxs[b * T_LEN + oi] : 0;
    }
    __syncthreads();

    // Per-thread staging geometry (independent of kc).
    int rA[4], cA[4];
    #pragma unroll
    for (int it = 0; it < 4; ++it) {
        int i = tid + it * 256;
        rA[it] = i >> 4;               // A row within tile (0..63)
        cA[it] = i & 15;               // A col within chunk (0..15)
    }
    int rowB[8];
    const float* pB[8];
    #pragma unroll
    for (int it = 0; it < 8; ++it) {
        int i    = tid + it * 256;
        rowB[it] = i >> 7;             // 0..15
        int col  = i & 127;
        pB[it]   = W1 + (long)rowB[it] * H_DIM + nBlk + col;   // advance by 16 rows per kc
    }

    float aReg[4], bReg[8];

    // ---- prologue: load chunk 0 into registers ----
    #pragma unroll
    for (int it = 0; it < 4; ++it) {
        int k = cA[it];                // kc = 0
        int kw = k / 300;
        int e  = k - kw * 300;
        int token = tokLds[rA[it] * 5 + kw];
        aReg[it] = emb[(long)token * E_DIM + e];
    }
    #pragma unroll
    for (int it = 0; it < 8; ++it) bReg[it] = pB[it][0];

    v8f acc00 = {}, acc01 = {}, acc10 = {}, acc11 = {};

    for (int kc = 0; kc < NUM_KC; ++kc) {
        // ---- commit staged registers to LDS ----
        #pragma unroll
        for (int it = 0; it < 4; ++it) Alds[rA[it] * 17 + cA[it]] = aReg[it];
        #pragma unroll
        for (int it = 0; it < 8; ++it) {
            int i = tid + it * 256;
            Blds[(i >> 7) * N_TILE + (i & 127)] = bReg[it];
        }
        __syncthreads();

        // ---- prefetch next chunk into registers (overlaps WMMAs below) ----
        if (kc + 1 < NUM_KC) {
            int kbase = (kc + 1) * K_CHUNK;
            #pragma unroll
            for (int it = 0; it < 4; ++it) {
                int k = kbase + cA[it];
                float v = 0.0f;
                if (k < K_WIN) {
                    int kw = k / 300;
                    int e  = k - kw * 300;
                    int token = tokLds[rA[it] * 5 + kw];
                    v = emb[(long)token * E_DIM + e];
                }
                aReg[it] = v;
            }
            #pragma unroll
            for (int it = 0; it < 8; ++it) {
                pB[it] += (long)K_CHUNK * H_DIM;
                bReg[it] = (kbase + rowB[it] < K_WIN) ? pB[it][0] : 0.0f;
            }
        }

        // ---- 16 WMMAs on the current LDS chunk ----
        #pragma unroll
        for (int kk = 0; kk < K_CHUNK; kk += 4) {
            int kA = kk + 2 * half;
            v2f a0, a1, b0, b1v;
            a0.x  = Alds[(waveM * 32 + l16) * 17 + kA];
            a0.y  = Alds[(waveM * 32 + l16) * 17 + kA + 1];
            a1.x  = Alds[(waveM * 32 + 16 + l16) * 17 + kA];
            a1.y  = Alds[(waveM * 32 + 16 + l16) * 17 + kA + 1];
            b0.x  = Blds[kA * N_TILE + waveN * 32 + l16];
            b0.y  = Blds[(kA + 1) * N_TILE + waveN * 32 + l16];
            b1v.x = Blds[kA * N_TILE + waveN * 32 + 16 + l16];
            b1v.y = Blds[(kA + 1) * N_TILE + waveN * 32 + 16 + l16];
            acc00 = __builtin_amdgcn_wmma_f32_16x16x4_f32(false, a0, false, b0,  (short)0, acc00, false, false);
            acc01 = __builtin_amdgcn_wmma_f32_16x16x4_f32(false, a0, false, b1v, (short)0, acc01, false, false);
            acc10 = __builtin_amdgcn_wmma_f32_16x16x4_f32(false, a1, false, b0,  (short)0, acc10, false, false);
            acc11 = __builtin_amdgcn_wmma_f32_16x16x4_f32(false, a1, false, b1v, (short)0, acc11, false, false);
        }
        __syncthreads();
    }

    // Store: C layout = 8 VGPRs, lanes 0-15 -> M=j, lanes 16-31 -> M=j+8.
    #pragma unroll
    for (int j = 0; j < 8; ++j) {
        int m0 = mBlk + waveM * 32 + j + 8 * half;
        int n0 = nBlk + waveN * 32 + l16;
        Hpre[(long)m0 * H_DIM + n0]              = acc00[j];
        Hpre[(long)m0 * H_DIM + n0 + 16]         = acc01[j];
        Hpre[(long)(m0 + 16) * H_DIM + n0]       = acc10[j];
        Hpre[(long)(m0 + 16) * H_DIM + n0 + 16]  = acc11[j];
    }
}

// ---------------------------------------------------------------------------
// Kernel 2: per-batch sequential scan. One block per b, 1024 threads (32
// waves), each thread owns 2 hidden units; weights held in registers.
// ---------------------------------------------------------------------------
__global__ __launch_bounds__(1024)
void seq_scan(const float* __restrict__ Hpre,
              const float* __restrict__ W1,
              const float* __restrict__ b1,
              const float* __restrict__ W2,
              const float* __restrict__ b2,
              const float* __restrict__ init_state,
              float* __restrict__ out) {
    __shared__ float sWred[32][5];
    __shared__ float sPrev[10];

    const int tid  = threadIdx.x;
    const int lane = tid & 31;
    const int wave = tid >> 5;
    const int b    = blockIdx.x;
    const int h0   = tid * 2;

    // Register-resident weights for this thread's two hidden units.
    float w1r[10][2];
    #pragma unroll
    for (int q = 0; q < 10; ++q) {
        w1r[q][0] = W1[(long)(K_WIN + q) * H_DIM + h0];
        w1r[q][1] = W1[(long)(K_WIN + q) * H_DIM + h0 + 1];
    }
    float w2r[2][5];
    #pragma unroll
    for (int j = 0; j < 2; ++j)
        #pragma unroll
        for (int o = 0; o < 5; ++o)
            w2r[j][o] = W2[(long)(h0 + j) * 5 + o];
    float b1r0 = b1[h0], b1r1 = b1[h0 + 1];
    float b2r[5];
    #pragma unroll
    for (int o = 0; o < 5; ++o) b2r[o] = b2[o];

    if (tid < 10) sPrev[tid] = init_state[tid];
    __syncthreads();

    const float* gbase = Hpre + (long)b * T_LEN * H_DIM;
    float gx = gbase[h0];
    float gy = gbase[h0 + 1];

    for (int t = 0; t < T_LEN; ++t) {
        // Prefetch next step's precomputed row (latency hidden by reduction).
        float gnx = 0.0f, gny = 0.0f;
        if (t + 1 < T_LEN) {
            const float* gp = gbase + (long)(t + 1) * H_DIM + h0;
            gnx = gp[0];
            gny = gp[1];
        }

        float p[10];
        #pragma unroll
        for (int q = 0; q < 10; ++q) p[q] = sPrev[q];

        float hv0 = gx + b1r0;
        float hv1 = gy + b1r1;
        #pragma unroll
        for (int q = 0; q < 10; ++q) {
            hv0 = fmaf(p[q], w1r[q][0], hv0);
            hv1 = fmaf(p[q], w1r[q][1], hv1);
        }
        hv0 = hv0 > 0.0f ? hv0 : 0.0f;
        hv1 = hv1 > 0.0f ? hv1 : 0.0f;

        float acc[5];
        #pragma unroll
        for (int o = 0; o < 5; ++o)
            acc[o] = hv0 * w2r[0][o] + hv1 * w2r[1][o];

        // wave32 butterfly reduction
        #pragma unroll
        for (int o = 0; o < 5; ++o) {
            #pragma unroll
            for (int off = 16; off > 0; off >>= 1)
                acc[o] += __shfl_xor(acc[o], off, 32);
        }
        if (lane == 0) {
            #pragma unroll
            for (int o = 0; o < 5; ++o) sWred[wave][o] = acc[o];
        }
        __syncthreads();

        if (tid == 0) {
            float y[5];
            #pragma unroll
            for (int o = 0; o < 5; ++o) {
                float s = b2r[o];
                for (int w = 0; w < 32; ++w) s += sWred[w][o];
                y[o] = s;
            }
            float* op = out + ((long)b * T_LEN + t) * 5;
            #pragma unroll
            for (int o = 0; o < 5; ++o) op[o] = y[o];
            // softmax -> shift prev
            float mx = y[0];
            #pragma unroll
            for (int o = 1; o < 5; ++o) mx = y[o] > mx ? y[o] : mx;
            float ex[5], s = 0.0f;
            #pragma unroll
            for (int o = 0; o < 5; ++o) { ex[o] = expf(y[o] - mx); s += ex[o]; }
            float inv = 1.0f / s;
            float s5 = sPrev[5], s6 = sPrev[6], s7 = sPrev[7], s8 = sPrev[8], s9 = sPrev[9];
            sPrev[0] = s5; sPrev[1] = s6; sPrev[2] = s7; sPrev[3] = s8; sPrev[4] = s9;
            #pragma unroll
            for (int o = 0; o < 5; ++o) sPrev[5 + o] = ex[o] * inv;
        }
        __syncthreads();

        gx = gnx;
        gy = gny;
    }
}

// ---------------------------------------------------------------------------
extern "C" void kernel_launch(void* const* d_in, const int* in_sizes, int n_in,
                              void* d_out, int out_size, void* d_ws, size_t ws_size,
                              hipStream_t stream) {
    const int*   idxs = (const int*)  d_in[0];   // [64, 512]
    const float* emb  = (const float*)d_in[1];   // [50000, 300]
    const float* W1   = (const float*)d_in[2];   // [1510, 2048]
    const float* b1   = (const float*)d_in[3];   // [2048]
    const float* W2   = (const float*)d_in[4];   // [2048, 5]
    const float* b2   = (const float*)d_in[5];   // [5]
    const float* init = (const float*)d_in[6];   // [10]
    float* out  = (float*)d_out;                 // [64, 512, 5]
    float* Hpre = (float*)d_ws;                  // [64*512, 2048] fp32 = 268 MB

    dim3 g1((64 * T_LEN) / M_TILE, H_DIM / N_TILE);   // (512, 16)
    winproj_gemm_f32<<<g1, 256, 0, stream>>>(idxs, emb, W1, Hpre);
    seq_scan<<<64, 1024, 0, stream>>>(Hpre, W1, b1, W2, b2, init, out);
}